// ESGCN_31662498906810
// MI455X (gfx1250) — compile-verified
//
#include <hip/hip_runtime.h>
#include <hip/hip_bf16.h>
#include <math.h>

// ---------------------------------------------------------------------------
// Problem constants (from reference): B=8, N=2048, T=8, HID=4, OUT=12
// ---------------------------------------------------------------------------
#define BB 8
#define NN 2048
#define TT 8
#define HID 4
#define OUTC 12

typedef __attribute__((ext_vector_type(2))) float v2f;
typedef __attribute__((ext_vector_type(8))) float v8f;

// Workspace layout (float offsets)
#define OFF_C0 0u          // cats0: 8*4*2048*8 = 524288
#define OFF_C1 524288u     // cats1: 8*4*2048*4 = 262144
#define OFF_C2 786432u     // cats2: 8*4*2048*2 = 131072
#define OFF_C3 917504u     // cats3 (F4): 8*4*2048 = 65536
#define OFF_FC 983040u     // Fc/Fl: 8*2048 = 16384
#define OFF_MX 999424u     // max_c F4: 16384
#define OFF_MN 1015808u    // min_c F4: 16384
#define OFF_L2 1032192u    // l2: 8  (+ pad to keep 16B alignment of OFF_X)
#define OFF_X  1032208u    // X: 8*2048*4 = 65536
// total ~= 1.1M floats = ~4.4 MB

// ---------------------------------------------------------------------------
// Fast transcendentals: gfx1250 V_TANH_F32 (confirmed in codegen); branchless
// exp/rcp fallback if the builtin ever disappears.
// ---------------------------------------------------------------------------
__device__ __forceinline__ float fast_rcp(float x) {
    return __builtin_amdgcn_rcpf(x);
}

__device__ __forceinline__ float fast_tanh(float x) {
#if __has_builtin(__builtin_amdgcn_tanhf)
    return __builtin_amdgcn_tanhf(x);
#else
    float ax = fabsf(x);
    float e  = __expf(2.0f * ax);
    float t  = 1.0f - 2.0f * fast_rcp(e + 1.0f);
    return copysignf(t, x);
#endif
}

__device__ __forceinline__ float fast_sigmoid(float x) {
    return fast_rcp(1.0f + __expf(-x));
}

// Async global->LDS copy, 16 bytes per lane (CDNA5 ASYNCcnt path).
__device__ __forceinline__ void async_lds_b128(uint32_t lds_off, const float* gaddr) {
    asm volatile("global_load_async_to_lds_b128 %0, %1, off"
                 :: "v"(lds_off), "v"(gaddr) : "memory");
}
__device__ __forceinline__ void wait_asynccnt0() {
    asm volatile("s_wait_asynccnt 0" ::: "memory");
}

// ---------------------------------------------------------------------------
// Stage: gated conv (sigmoid*tanh) + per-batch global LayerNorm.
// One block per batch; global mean/var over all 4*2048*lenout elements.
// ---------------------------------------------------------------------------
__global__ void stage_kernel(const float* __restrict__ in,
                             const float* __restrict__ w1, const float* __restrict__ b1,
                             const float* __restrict__ w2, const float* __restrict__ b2,
                             const float* __restrict__ lng, const float* __restrict__ lnb,
                             float* __restrict__ out,
                             int ins, int lenin, int lenout, int stride, int from_x) {
    __shared__ float sw1[48], sw2[48], sb1[4], sb2[4];
    __shared__ float sred[512];
    __shared__ float sStats[2];

    const int b = blockIdx.x;
    const int tid = threadIdx.x;
    const int nw = 4 * ins * 3;
    if (tid < nw) { sw1[tid] = w1[tid]; sw2[tid] = w2[tid]; }
    if (tid < 4)  { sb1[tid] = b1[tid]; sb2[tid] = b2[tid]; }
    __syncthreads();

    const int total = 4 * NN * lenout;
    float lsum = 0.0f, lsq = 0.0f;

    for (int e = tid; e < total; e += 512) {
        int c = e / (NN * lenout);
        int rem = e - c * (NN * lenout);
        int n = rem / lenout;
        int t = rem - n * lenout;

        float z1 = sb1[c], z2 = sb2[c];
        #pragma unroll
        for (int dx = 0; dx < 3; ++dx) {
            int tp = t * stride + dx - 1;
            if (tp >= 0 && tp < lenin) {
                for (int i = 0; i < ins; ++i) {
                    float v = from_x ? in[(b * TT + tp) * NN + n]
                                     : in[((b * ins + i) * NN + n) * lenin + tp];
                    z1 = fmaf(sw1[(c * ins + i) * 3 + dx], v, z1);
                    z2 = fmaf(sw2[(c * ins + i) * 3 + dx], v, z2);
                }
            }
        }
        float h = fast_sigmoid(z1) * fast_tanh(z2);
        out[b * total + e] = h;
        lsum += h;
        lsq  += h * h;
    }

    // block reductions (fixed order -> deterministic)
    sred[tid] = lsum; __syncthreads();
    for (int s = 256; s > 0; s >>= 1) { if (tid < s) sred[tid] += sred[tid + s]; __syncthreads(); }
    if (tid == 0) sStats[0] = sred[0];
    __syncthreads();
    sred[tid] = lsq; __syncthreads();
    for (int s = 256; s > 0; s >>= 1) { if (tid < s) sred[tid] += sred[tid + s]; __syncthreads(); }
    if (tid == 0) sStats[1] = sred[0];
    __syncthreads();

    const float invc = 1.0f / (float)total;
    const float mean = sStats[0] * invc;
    const float var  = sStats[1] * invc - mean * mean;
    const float rstd = rsqrtf(var + 1e-5f);

    for (int e = tid; e < total; e += 512) {
        float h = out[b * total + e];
        // lng/lnb layout (4, N, lenout) row-major == e's (c,n,t) decode
        out[b * total + e] = (h - mean) * rstd * lng[e] + lnb[e];
    }
}

// ---------------------------------------------------------------------------
// Fc[b,n] = stcc_b + sum_c stcc_w[c]*F4[b,c,n]; also per-row max/min over c,
// and l2[b] = sum_n Fc^2   (xx==yy in the reference, so l2 = |Fl|^2 exactly)
// ---------------------------------------------------------------------------
__global__ void fc_kernel(const float* __restrict__ F4,
                          const float* __restrict__ stcc_w, const float* __restrict__ stcc_b,
                          float* __restrict__ Fc, float* __restrict__ mxv,
                          float* __restrict__ mnv, float* __restrict__ l2) {
    __shared__ float sred[256];
    const int b = blockIdx.x;
    const int tid = threadIdx.x;
    const float w0 = stcc_w[0], w1 = stcc_w[1], w2 = stcc_w[2], w3 = stcc_w[3];
    const float bc = stcc_b[0];
    float lsq = 0.0f;
    for (int n = tid; n < NN; n += 256) {
        float f0 = F4[(b * 4 + 0) * NN + n];
        float f1 = F4[(b * 4 + 1) * NN + n];
        float f2 = F4[(b * 4 + 2) * NN + n];
        float f3 = F4[(b * 4 + 3) * NN + n];
        float fc = bc + w0 * f0 + w1 * f1 + w2 * f2 + w3 * f3;
        Fc[b * NN + n]  = fc;
        mxv[b * NN + n] = fmaxf(fmaxf(f0, f1), fmaxf(f2, f3));
        mnv[b * NN + n] = fminf(fminf(f0, f1), fminf(f2, f3));
        lsq += fc * fc;
    }
    sred[tid] = lsq; __syncthreads();
    for (int s = 128; s > 0; s >>= 1) { if (tid < s) sred[tid] += sred[tid + s]; __syncthreads(); }
    if (tid == 0) l2[b] = sred[0];
}

// ---------------------------------------------------------------------------
// X[b,k,c] = sum_m W[b,k,m] * F4[b,c,m]
//   u = Fl[k]*Fl[m]/l2 ;  e = (u>=0) ? max_c F4[:,k] : min_c F4[:,k]
//   W = u * relu(tanh(u*e))
// Accumulated with V_WMMA_F32_16X16X4_F32: D(16k x 16n, n<4 used) += A(16x4)B(4x16)
// One wave per 16-row k tile; 8 waves per block; Fl & F4 staged in LDS via
// the CDNA5 async global->LDS path (ASYNCcnt). sF4 carries a zero-filled
// padding row (index 4) so lanes r>=4 issue unconditional ds_load_b64 from
// a zero row instead of exec-predicated loads.
// ---------------------------------------------------------------------------
__global__ void x_wmma_kernel(const float* __restrict__ Fl, const float* __restrict__ l2,
                              const float* __restrict__ mxv, const float* __restrict__ mnv,
                              const float* __restrict__ F4, float* __restrict__ X) {
    __shared__ __align__(16) float sFl[NN];
    __shared__ __align__(16) float sF4[5 * NN];   // rows 0..3 = channels, row 4 = zeros

    const int b = blockIdx.y;

    // Async-stage Fl (8KB) and F4 (32KB) into LDS, 16B per lane per issue;
    // zero the padding row with plain DS stores in parallel.
    {
        const uint32_t flBase = (uint32_t)(size_t)(void*)sFl;   // flat->LDS: low 32 bits
        const uint32_t f4Base = (uint32_t)(size_t)(void*)sF4;
        for (int i = threadIdx.x * 4; i < NN; i += 256 * 4)
            async_lds_b128(flBase + (uint32_t)(i * 4), Fl + (size_t)b * NN + i);
        for (int i = threadIdx.x * 4; i < 4 * NN; i += 256 * 4)
            async_lds_b128(f4Base + (uint32_t)(i * 4), F4 + (size_t)b * 4 * NN + i);
        for (int i = threadIdx.x; i < NN; i += 256)
            sF4[4 * NN + i] = 0.0f;
        wait_asynccnt0();
    }
    __syncthreads();

    const int wave = threadIdx.x >> 5;          // 0..7
    const int lane = threadIdx.x & 31;
    const int ktile = blockIdx.x * 8 + wave;    // 0..127
    const int r = lane & 15;                    // A row (k) / B column (c)
    const int half = lane >> 4;

    const int kg = ktile * 16 + r;
    const float inv = 1.0f / l2[b];
    const float qk = sFl[kg] * inv;             // Fl[k]/l2
    const float ep = mxv[b * NN + kg];
    const float en = mnv[b * NN + kg];
    const int mbase = half * 2;                 // A/B f32 K-layout: vgpr0->{K0,K2}, vgpr1->{K1,K3}
    const bool cvalid = (r < 4);
    // lanes with r>=4 read the zero-filled padding row -> B columns 4..15 are 0
    const float* bptr = &sF4[(cvalid ? r : 4) * NN];

    v8f acc = {};
    #pragma unroll 2
    for (int m0 = 0; m0 < NN; m0 += 4) {
        const int mx = m0 + mbase;
        const int my = mx + 1;
        const float ux = qk * sFl[mx];
        const float uy = qk * sFl[my];
        const float ex = (ux >= 0.0f) ? ep : en;
        const float ey = (uy >= 0.0f) ? ep : en;
        const float ax = fmaxf(fast_tanh(ux * ex), 0.0f);
        const float ay = fmaxf(fast_tanh(uy * ey), 0.0f);
        v2f A; A.x = ux * ax; A.y = uy * ay;
        v2f Bv; Bv.x = bptr[mx]; Bv.y = bptr[my];
        acc = __builtin_amdgcn_wmma_f32_16x16x4_f32(
            /*neg_a=*/false, A, /*neg_b=*/false, Bv,
            /*c_mod=*/(short)0, acc, /*reuse_a=*/false, /*reuse_b=*/false);
    }

    // D layout: vgpr j, lanes 0-15 -> M=j,N=lane ; lanes 16-31 -> M=j+8,N=lane-16
    if (cvalid) {
        #pragma unroll
        for (int v = 0; v < 8; ++v) {
            int krow = ktile * 16 + v + 8 * half;
            X[(b * NN + krow) * 4 + r] = acc[v];
        }
    }
}

// ---------------------------------------------------------------------------
// Downstream small algebra per (b,n): gcn, ce, ci-fusion (concat/add), fc1+relu, fc2
// ---------------------------------------------------------------------------
__global__ void final_kernel(const float* __restrict__ X,
                             const float* __restrict__ gcn_w, const float* __restrict__ gcn_b,
                             const float* __restrict__ ce_w, const float* __restrict__ ce_b,
                             const float* __restrict__ ci_w, const float* __restrict__ ci_b,
                             const float* __restrict__ fc1_w, const float* __restrict__ fc1_b,
                             const float* __restrict__ fc2_w, const float* __restrict__ fc2_b,
                             const float* __restrict__ c0, const float* __restrict__ c1,
                             const float* __restrict__ c2, const float* __restrict__ c3,
                             float* __restrict__ y) {
    const int g = blockIdx.x * 256 + threadIdx.x;      // b*N + n
    if (g >= BB * NN) return;
    const int b = g >> 11;
    const int n = g & (NN - 1);

    float Xv[4];
    #pragma unroll
    for (int c = 0; c < 4; ++c) Xv[c] = X[g * 4 + c];

    float Fg[4];
    #pragma unroll
    for (int d = 0; d < 4; ++d) {
        float a = gcn_b[n * 4 + d];
        #pragma unroll
        for (int c = 0; c < 4; ++c) a = fmaf(gcn_w[(n * 4 + d) * 4 + c], Xv[c], a);
        Fg[d] = a;
    }
    float ce[4];
    #pragma unroll
    for (int o = 0; o < 4; ++o) {
        float a = ce_b[o];
        #pragma unroll
        for (int d = 0; d < 4; ++d) a = fmaf(ce_w[o * 4 + d], Fg[d], a);
        ce[o] = a;
    }

    // ci residuals: ciK[o][t'] = ci_b[K,o] + sum_c ci_w[K,o,c]*catsK[b,c,n,t']
    float ci0[4][8], ci1[4][4], ci2[4][2], ci3[4];
    #pragma unroll
    for (int o = 0; o < 4; ++o) {
        #pragma unroll
        for (int t = 0; t < 8; ++t) ci0[o][t] = ci_b[0 * 4 + o];
        #pragma unroll
        for (int t = 0; t < 4; ++t) ci1[o][t] = ci_b[1 * 4 + o];
        #pragma unroll
        for (int t = 0; t < 2; ++t) ci2[o][t] = ci_b[2 * 4 + o];
        ci3[o] = ci_b[3 * 4 + o];
    }
    #pragma unroll
    for (int c = 0; c < 4; ++c) {
        float v0[8], v1[4], v2[2], v3;
        #pragma unroll
        for (int t = 0; t < 8; ++t) v0[t] = c0[((b * 4 + c) * NN + n) * 8 + t];
        #pragma unroll
        for (int t = 0; t < 4; ++t) v1[t] = c1[((b * 4 + c) * NN + n) * 4 + t];
        #pragma unroll
        for (int t = 0; t < 2; ++t) v2[t] = c2[((b * 4 + c) * NN + n) * 2 + t];
        v3 = c3[(b * 4 + c) * NN + n];
        #pragma unroll
        for (int o = 0; o < 4; ++o) {
            const float w0 = ci_w[(0 * 4 + o) * 4 + c];
            const float w1 = ci_w[(1 * 4 + o) * 4 + c];
            const float w2 = ci_w[(2 * 4 + o) * 4 + c];
            const float w3 = ci_w[(3 * 4 + o) * 4 + c];
            #pragma unroll
            for (int t = 0; t < 8; ++t) ci0[o][t] = fmaf(w0, v0[t], ci0[o][t]);
            #pragma unroll
            for (int t = 0; t < 4; ++t) ci1[o][t] = fmaf(w1, v1[t], ci1[o][t]);
            #pragma unroll
            for (int t = 0; t < 2; ++t) ci2[o][t] = fmaf(w2, v2[t], ci2[o][t]);
            ci3[o] = fmaf(w3, v3, ci3[o]);
        }
    }

    // concat order: [ce, ci3, ci2(2), ci1(4)] then += ci0 elementwise (width 8)
    float outv[4][8];
    #pragma unroll
    for (int o = 0; o < 4; ++o) {
        outv[o][0] = ce[o]      + ci0[o][0];
        outv[o][1] = ci3[o]     + ci0[o][1];
        outv[o][2] = ci2[o][0]  + ci0[o][2];
        outv[o][3] = ci2[o][1]  + ci0[o][3];
        #pragma unroll
        for (int t = 4; t < 8; ++t) outv[o][t] = ci1[o][t - 4] + ci0[o][t];
    }

    #pragma unroll
    for (int t = 0; t < 8; ++t) {
        float h[4];
        #pragma unroll
        for (int o = 0; o < 4; ++o) {
            float a = fc1_b[o];
            #pragma unroll
            for (int c = 0; c < 4; ++c) a = fmaf(fc1_w[o * 4 + c], outv[c][t], a);
            h[o] = fmaxf(a, 0.0f);
        }
        const size_t base = ((size_t)(b * TT + t) * NN + n) * OUTC;
        #pragma unroll
        for (int j = 0; j < OUTC; ++j) {
            float a = fc2_b[j];
            #pragma unroll
            for (int o = 0; o < 4; ++o) a = fmaf(fc2_w[j * 4 + o], h[o], a);
            y[base + j] = a;
        }
    }
}

// ---------------------------------------------------------------------------
extern "C" void kernel_launch(void* const* d_in, const int* in_sizes, int n_in,
                              void* d_out, int out_size, void* d_ws, size_t ws_size,
                              hipStream_t stream) {
    (void)in_sizes; (void)n_in; (void)out_size; (void)ws_size;
    const float* x = (const float*)d_in[0];
    // per-stage params: [1 + 6*s + j]
    const float* s_c1w[4]; const float* s_c1b[4];
    const float* s_c2w[4]; const float* s_c2b[4];
    const float* s_lng[4]; const float* s_lnb[4];
    for (int s = 0; s < 4; ++s) {
        s_c1w[s] = (const float*)d_in[1 + 6 * s + 0];
        s_c1b[s] = (const float*)d_in[1 + 6 * s + 1];
        s_c2w[s] = (const float*)d_in[1 + 6 * s + 2];
        s_c2b[s] = (const float*)d_in[1 + 6 * s + 3];
        s_lng[s] = (const float*)d_in[1 + 6 * s + 4];
        s_lnb[s] = (const float*)d_in[1 + 6 * s + 5];
    }
    const float* stcc_w = (const float*)d_in[25];
    const float* stcc_b = (const float*)d_in[26];
    const float* gcn_w  = (const float*)d_in[27];
    const float* gcn_b  = (const float*)d_in[28];
    const float* ce_w   = (const float*)d_in[29];
    const float* ce_b   = (const float*)d_in[30];
    const float* ci_w   = (const float*)d_in[31];
    const float* ci_b   = (const float*)d_in[32];
    const float* fc1_w  = (const float*)d_in[33];
    const float* fc1_b  = (const float*)d_in[34];
    const float* fc2_w  = (const float*)d_in[35];
    const float* fc2_b  = (const float*)d_in[36];

    float* ws = (float*)d_ws;
    float* cats[4] = { ws + OFF_C0, ws + OFF_C1, ws + OFF_C2, ws + OFF_C3 };
    float* Fc  = ws + OFF_FC;
    float* mxv = ws + OFF_MX;
    float* mnv = ws + OFF_MN;
    float* l2  = ws + OFF_L2;
    float* X   = ws + OFF_X;
    float* y   = (float*)d_out;

    // stage params: ins, lenin, lenout, stride
    const int insA[4]    = {1, 4, 4, 4};
    const int leninA[4]  = {8, 8, 4, 2};
    const int lenoutA[4] = {8, 4, 2, 1};
    const int strideA[4] = {1, 2, 2, 2};
    for (int s = 0; s < 4; ++s) {
        const float* in = (s == 0) ? x : cats[s - 1];
        stage_kernel<<<BB, 512, 0, stream>>>(in, s_c1w[s], s_c1b[s], s_c2w[s], s_c2b[s],
                                             s_lng[s], s_lnb[s], cats[s],
                                             insA[s], leninA[s], lenoutA[s], strideA[s],
                                             (s == 0) ? 1 : 0);
    }

    fc_kernel<<<BB, 256, 0, stream>>>(cats[3], stcc_w, stcc_b, Fc, mxv, mnv, l2);

    // 128 k-tiles of 16 rows; 8 waves (tiles) per block -> grid.x = 16, per batch
    x_wmma_kernel<<<dim3(16, BB), 256, 0, stream>>>(Fc, l2, mxv, mnv, cats[3], X);

    final_kernel<<<(BB * NN + 255) / 256, 256, 0, stream>>>(
        X, gcn_w, gcn_b, ce_w, ce_b, ci_w, ci_b, fc1_w, fc1_b, fc2_w, fc2_b,
        cats[0], cats[1], cats[2], cats[3], y);
}